// GATEncoder_59957743452469
// MI455X (gfx1250) — compile-verified
//
#include <hip/hip_runtime.h>
#include <hip/hip_bf16.h>

// ---------------------------------------------------------------------------
// GAT encoder for MI455X (gfx1250, wave32).
//   Layer1: GATConv(128 -> 8 heads x 64) + ReLU
//   Layer2: GATConv(512 -> 1 head  x 32)
//   Output: (mu, logstd) = split of final 32 cols.
// Dense GEMMs use V_WMMA_F32_16X16X4_F32 (fp32 tensor path, one wave per
// 16x16 output tile). Edge softmax-aggregation uses L2 fp32 atomics.
// ---------------------------------------------------------------------------

#define NNODES 10000
#define HEADS1 8
#define CH1    64
#define F1     (HEADS1 * CH1)   // 512
#define CH2    32
#define NEGSLOPE 0.2f

typedef __attribute__((ext_vector_type(2))) float v2f;
typedef __attribute__((ext_vector_type(8))) float v8f;

// ---- monotone float <-> uint encoding for atomicMax on floats --------------
__device__ __forceinline__ unsigned enc_f32(float f) {
    unsigned u = __float_as_uint(f);
    return (u & 0x80000000u) ? ~u : (u | 0x80000000u);
}
__device__ __forceinline__ float dec_f32(unsigned u) {
    return (u & 0x80000000u) ? __uint_as_float(u & 0x7fffffffu)
                             : __uint_as_float(~u);
}
__device__ __forceinline__ float lrelu(float v) {
    return v > 0.0f ? v : NEGSLOPE * v;
}

// ---- zero init -------------------------------------------------------------
__global__ void k_zero(float* __restrict__ p, size_t count) {
    size_t i = (size_t)blockIdx.x * blockDim.x + threadIdx.x;
    if (i < count) p[i] = 0.0f;
}

// ---- fp32 WMMA GEMM: C[M,N] = A[M,K] * B[K,N]; one wave per 16x16 tile -----
// A fragment (16x4 f32): VGPR0 = {K=k0 | K=k0+2}, VGPR1 = {K=k0+1 | K=k0+3}
//   lane 0-15 -> M=lane, lane 16-31 -> M=lane-16, K shifted by +2.
// B fragment (4x16 f32): VGPR0 = {K=k0 | K=k0+2}, VGPR1 = {K=k0+1 | K=k0+3},
//   N = lane & 15.
// C/D (16x16 f32): VGPR v, lanes 0-15 -> M=v, lanes 16-31 -> M=v+8, N=lane&15.
__global__ void k_gemm_wmma_f32(const float* __restrict__ A,
                                const float* __restrict__ B,
                                float* __restrict__ C,
                                int M, int K, int N) {
    int wave = (int)((blockIdx.x * blockDim.x + threadIdx.x) >> 5);
    int lane = threadIdx.x & 31;
    int tiles_n = N >> 4;
    int tiles   = (M >> 4) * tiles_n;
    if (wave >= tiles) return;
    int tm = wave / tiles_n;
    int tn = wave - tm * tiles_n;
    int row0 = tm << 4, col0 = tn << 4;

    int mh   = lane & 15;           // M (for A) / N (for B,C)
    int koff = (lane >> 4) << 1;    // 0 for lanes 0-15, 2 for lanes 16-31

    v8f acc = {};
    const float* arow = A + (size_t)(row0 + mh) * K;
    for (int k0 = 0; k0 < K; k0 += 4) {
        v2f a, b;
        const float* ap = arow + k0 + koff;
        a.x = ap[0];
        a.y = ap[1];
        const float* bp = B + (size_t)(k0 + koff) * N + col0 + mh;
        b.x = bp[0];
        b.y = bp[N];
        acc = __builtin_amdgcn_wmma_f32_16x16x4_f32(
            /*neg_a=*/false, a, /*neg_b=*/false, b,
            /*c_mod=*/(short)0, acc, /*reuse_a=*/false, /*reuse_b=*/false);
    }
    float* cp = C + (size_t)(row0 + ((lane >> 4) << 3)) * N + col0 + mh;
#pragma unroll
    for (int v = 0; v < 8; ++v) cp[(size_t)v * N] = acc[v];
}

// ---- per-node attention logits: asrc/adst[n,h] = <h[n,h,:], a_src/dst[h,:]>
__global__ void k_alpha(const float* __restrict__ h,
                        const float* __restrict__ a_src,
                        const float* __restrict__ a_dst,
                        float* __restrict__ asrc, float* __restrict__ adst,
                        int n, int heads, int ch) {
    int i = blockIdx.x * blockDim.x + threadIdx.x;
    if (i >= n * heads) return;
    int node = i / heads, hd = i - node * heads;
    const float* hp = h + (size_t)node * heads * ch + (size_t)hd * ch;
    const float* as = a_src + hd * ch;
    const float* ad = a_dst + hd * ch;
    float s = 0.0f, d = 0.0f;
    for (int c = 0; c < ch; ++c) {
        float v = hp[c];
        s += v * as[c];
        d += v * ad[c];
    }
    asrc[i] = s;
    adst[i] = d;
}

// ---- edge pass 1: segment max of leaky-relu logits over dst ---------------
__global__ void k_edge_max(const int* __restrict__ src,
                           const int* __restrict__ dst,
                           const float* __restrict__ asrc,
                           const float* __restrict__ adst,
                           unsigned* __restrict__ emax,
                           int EA, int E, int heads) {
    int i = blockIdx.x * blockDim.x + threadIdx.x;
    if (i >= EA * heads) return;
    int e = i / heads, hd = i - e * heads;
    int s = (e < E) ? src[e] : (e - E);   // appended self-loops
    int d = (e < E) ? dst[e] : (e - E);
    float v = lrelu(asrc[s * heads + hd] + adst[d * heads + hd]);
    atomicMax(&emax[d * heads + hd], enc_f32(v));
}

// ---- edge pass 2: denom[dst,h] += ee ; agg[dst,:] += ee * h[src,:] --------
// One wave (32 lanes) per edge; ee computed once per head, shfl-broadcast.
__global__ void k_edge_agg(const int* __restrict__ src,
                           const int* __restrict__ dst,
                           const float* __restrict__ asrc,
                           const float* __restrict__ adst,
                           const unsigned* __restrict__ emax,
                           float* __restrict__ denom,
                           float* __restrict__ agg,
                           const float* __restrict__ h,
                           int EA, int E, int heads, int ch) {
    int wave = (int)((blockIdx.x * blockDim.x + threadIdx.x) >> 5);
    int lane = threadIdx.x & 31;
    if (wave >= EA) return;
    int e = wave;
    int s = (e < E) ? src[e] : (e - E);
    int d = (e < E) ? dst[e] : (e - E);
    int F = heads * ch;

    float ee_l = 0.0f;
    if (lane < heads) {
        float v = lrelu(asrc[s * heads + lane] + adst[d * heads + lane]);
        ee_l = __expf(v - dec_f32(emax[d * heads + lane]));
        atomicAdd(&denom[d * heads + lane], ee_l);
    }
    const float* hs = h + (size_t)s * F;
    float* ag = agg + (size_t)d * F;
    for (int f = lane; f < F; f += 32) {
        int hd = f / ch;
        float ee = __shfl(ee_l, hd);
        atomicAdd(&ag[f], ee * hs[f]);
    }
}

// ---- normalize + bias + relu (layer-1 epilogue, in place) -----------------
__global__ void k_norm_bias_relu(float* __restrict__ agg,
                                 const float* __restrict__ denom,
                                 const float* __restrict__ b,
                                 int n, int heads, int ch) {
    int F = heads * ch;
    int i = blockIdx.x * blockDim.x + threadIdx.x;
    if (i >= n * F) return;
    int node = i / F, f = i - node * F;
    float v = agg[i] / (denom[node * heads + f / ch] + 1e-16f) + b[f];
    agg[i] = v > 0.0f ? v : 0.0f;
}

// ---- layer-2 epilogue: normalize + bias, split into (mu, logstd) ----------
__global__ void k_final(const float* __restrict__ agg,
                        const float* __restrict__ denom,
                        const float* __restrict__ b,
                        float* __restrict__ out, int n) {
    int i = blockIdx.x * blockDim.x + threadIdx.x;
    if (i >= n * CH2) return;
    int node = i >> 5, f = i & (CH2 - 1);
    float v = agg[i] / (denom[node] + 1e-16f) + b[f];
    int half = CH2 / 2;                 // 16
    if (f < half) out[(size_t)node * half + f] = v;
    else          out[(size_t)n * half + (size_t)node * half + (f - half)] = v;
}

// ---------------------------------------------------------------------------
extern "C" void kernel_launch(void* const* d_in, const int* in_sizes, int n_in,
                              void* d_out, int out_size, void* d_ws, size_t ws_size,
                              hipStream_t stream) {
    const float* x      = (const float*)d_in[0];
    const int*   eidx   = (const int*)d_in[1];
    const float* W1     = (const float*)d_in[2];
    const float* a_src1 = (const float*)d_in[3];
    const float* a_dst1 = (const float*)d_in[4];
    const float* b1     = (const float*)d_in[5];
    const float* W2     = (const float*)d_in[6];
    const float* a_src2 = (const float*)d_in[7];
    const float* a_dst2 = (const float*)d_in[8];
    const float* b2     = (const float*)d_in[9];
    float* out = (float*)d_out;

    const int N  = NNODES;
    const int E  = in_sizes[1] / 2;
    const int EA = E + N;                     // edges + self-loops
    const int IN = in_sizes[0] / N;           // 128
    const int* srcI = eidx;
    const int* dstI = eidx + E;

    // Workspace layout (floats). Zero-initialized block first, contiguous.
    float* w = (float*)d_ws;
    size_t off = 0;
    float*    agg1   = w + off; off += (size_t)N * F1;   // zeroed
    float*    denom1 = w + off; off += (size_t)N * HEADS1;
    unsigned* emax1  = (unsigned*)(w + off); off += (size_t)N * HEADS1;
    float*    agg2   = w + off; off += (size_t)N * CH2;
    float*    denom2 = w + off; off += (size_t)N;
    unsigned* emax2  = (unsigned*)(w + off); off += (size_t)N;
    size_t zcount = off;                                  // end of zero block
    float*    h1     = w + off; off += (size_t)N * F1;
    float*    asrc1  = w + off; off += (size_t)N * HEADS1;
    float*    adst1  = w + off; off += (size_t)N * HEADS1;
    float*    h2     = w + off; off += (size_t)N * CH2;
    float*    asrc2  = w + off; off += (size_t)N;
    float*    adst2  = w + off; off += (size_t)N;
    (void)ws_size; (void)n_in; (void)out_size;

    const int BLK = 256;
    auto cdiv = [](size_t a, size_t b) { return (int)((a + b - 1) / b); };

    // 0) zero accumulators (emax encoding: 0 < enc(-inf), self-loops cover all)
    k_zero<<<cdiv(zcount, BLK), BLK, 0, stream>>>(w, zcount);

    // 1) h1 = x @ W1   (10000x128 * 128x512), WMMA fp32
    {
        int waves = (N / 16) * (F1 / 16);
        k_gemm_wmma_f32<<<cdiv((size_t)waves * 32, BLK), BLK, 0, stream>>>(
            x, W1, h1, N, IN, F1);
    }
    // 2) attention logits per node/head
    k_alpha<<<cdiv((size_t)N * HEADS1, BLK), BLK, 0, stream>>>(
        h1, a_src1, a_dst1, asrc1, adst1, N, HEADS1, CH1);
    // 3) segment max over dst
    k_edge_max<<<cdiv((size_t)EA * HEADS1, BLK), BLK, 0, stream>>>(
        srcI, dstI, asrc1, adst1, emax1, EA, E, HEADS1);
    // 4) weighted aggregation (wave per edge)
    k_edge_agg<<<cdiv((size_t)EA * 32, BLK), BLK, 0, stream>>>(
        srcI, dstI, asrc1, adst1, emax1, denom1, agg1, h1, EA, E, HEADS1, CH1);
    // 5) normalize + bias + relu -> layer-2 input (in place in agg1)
    k_norm_bias_relu<<<cdiv((size_t)N * F1, BLK), BLK, 0, stream>>>(
        agg1, denom1, b1, N, HEADS1, CH1);

    // 6) h2 = relu_h @ W2  (10000x512 * 512x32), WMMA fp32
    {
        int waves = (N / 16) * (CH2 / 16);
        k_gemm_wmma_f32<<<cdiv((size_t)waves * 32, BLK), BLK, 0, stream>>>(
            agg1, W2, h2, N, F1, CH2);
    }
    // 7) layer-2 attention (1 head, 32 ch)
    k_alpha<<<cdiv((size_t)N, BLK), BLK, 0, stream>>>(
        h2, a_src2, a_dst2, asrc2, adst2, N, 1, CH2);
    k_edge_max<<<cdiv((size_t)EA, BLK), BLK, 0, stream>>>(
        srcI, dstI, asrc2, adst2, emax2, EA, E, 1);
    k_edge_agg<<<cdiv((size_t)EA * 32, BLK), BLK, 0, stream>>>(
        srcI, dstI, asrc2, adst2, emax2, denom2, agg2, h2, EA, E, 1, CH2);
    // 8) normalize + bias, split into (mu, logstd)
    k_final<<<cdiv((size_t)N * CH2, BLK), BLK, 0, stream>>>(
        agg2, denom2, b2, out, N);
}